// CAMD_32349693673977
// MI455X (gfx1250) — compile-verified
//
#include <hip/hip_runtime.h>
#include <hip/hip_bf16.h>

typedef __attribute__((ext_vector_type(2))) float v2f;
typedef __attribute__((ext_vector_type(8))) float v8f;

static __device__ __forceinline__ v8f wmma_f32(v2f a, v2f b, v8f c) {
  // D = A(16x4) * B(4x16) + C(16x16), all fp32
  return __builtin_amdgcn_wmma_f32_16x16x4_f32(false, a, false, b, (short)0, c, false, false);
}

// gfx1250 async global->LDS copy (ASYNCcnt-tracked), 16 bytes per lane.
// lds_addr: wave-relative LDS byte address (low 32 bits of generic pointer).
static __device__ __forceinline__ void async_ld_lds_b128(unsigned lds_addr,
                                                         const void* gptr) {
  asm volatile("global_load_async_to_lds_b128 %0, %1, off"
               :: "v"(lds_addr), "v"((unsigned long long)(uintptr_t)gptr)
               : "memory");
}
static __device__ __forceinline__ void wait_asynccnt0() {
  asm volatile("s_wait_asynccnt 0x0" ::: "memory");
}

// ---------------------------------------------------------------------------
// Kernel A: fused 2-layer MLP  Y = act( relu(X@W1+b1) @ W2 + b2 )
// act==1 -> elu(y)+1  (Q and K feature map), act==0 -> identity (V)
// blockDim = 128 (4 waves), each wave owns one 16-row tile; grid = T/64
// ---------------------------------------------------------------------------
__global__ __launch_bounds__(128)
void CAMD_mlp_kernel(const float* __restrict__ X, int d_in,
                     const float* __restrict__ w1, const float* __restrict__ b1,
                     const float* __restrict__ w2, const float* __restrict__ b2,
                     float* __restrict__ Y, int act)
{
  __shared__ float lw1[64 * 64];
  __shared__ float lw2[64 * 64];
  __shared__ float lb1[64];
  __shared__ float lb2[64];
  __shared__ float Ht[4][16 * 64];

  const int tid = threadIdx.x;
  for (int i = tid; i < d_in * 64; i += 128) lw1[i] = w1[i];
  for (int i = tid; i < 64 * 64; i += 128)   lw2[i] = w2[i];
  if (tid < 64) { lb1[tid] = b1[tid]; lb2[tid] = b2[tid]; }
  __syncthreads();

  const int wave = tid >> 5;
  const int lane = tid & 31;
  const int col  = lane & 15;   // C-layout column / A-layout row
  const int hi   = lane >> 4;   // half-wave selector (K pair / row offset)
  const int row0 = blockIdx.x * 64 + wave * 16;
  const float* Xrow = X + (size_t)(row0 + col) * d_in;
  float* Hw = Ht[wave];

  const int nk = d_in >> 2;

  // Layer 1: H = relu(X @ W1 + b1)   (16 x 64 tile)
  #pragma unroll
  for (int nt = 0; nt < 4; ++nt) {
    const float bias = lb1[nt * 16 + col];
    v8f acc;
    #pragma unroll
    for (int v = 0; v < 8; ++v) acc[v] = bias;
    for (int kc = 0; kc < nk; ++kc) {
      const int k0 = kc * 4 + hi * 2;
      v2f a, b;
      a[0] = Xrow[k0];
      a[1] = Xrow[k0 + 1];
      b[0] = lw1[(k0)     * 64 + nt * 16 + col];
      b[1] = lw1[(k0 + 1) * 64 + nt * 16 + col];
      acc = wmma_f32(a, b, acc);
    }
    #pragma unroll
    for (int v = 0; v < 8; ++v) {
      float h = acc[v];
      Hw[(v + 8 * hi) * 64 + nt * 16 + col] = h > 0.f ? h : 0.f;
    }
  }
  __syncthreads();

  // Layer 2: Y = H @ W2 + b2, optional elu+1
  #pragma unroll
  for (int nt = 0; nt < 4; ++nt) {
    const float bias = lb2[nt * 16 + col];
    v8f acc;
    #pragma unroll
    for (int v = 0; v < 8; ++v) acc[v] = bias;
    #pragma unroll
    for (int kc = 0; kc < 16; ++kc) {
      const int k0 = kc * 4 + hi * 2;
      v2f a, b;
      a[0] = Hw[col * 64 + k0];
      a[1] = Hw[col * 64 + k0 + 1];
      b[0] = lw2[(k0)     * 64 + nt * 16 + col];
      b[1] = lw2[(k0 + 1) * 64 + nt * 16 + col];
      acc = wmma_f32(a, b, acc);
    }
    #pragma unroll
    for (int v = 0; v < 8; ++v) {
      float y = acc[v];
      if (act) y = (y > 0.f) ? (y + 1.f) : __expf(y);  // elu(y)+1
      Y[(size_t)(row0 + v + 8 * hi) * 64 + nt * 16 + col] = y;
    }
  }
}

// ---------------------------------------------------------------------------
// Kernel B: per-chunk KV state   KV_c[k][v] = sum_{t in chunk} K[t][k]*V[t][v]
// chunk = 64 keys; blockDim = 128 (4 waves, one 16-row band of k-dim each).
// K/V chunks are staged into LDS via gfx1250 async global->LDS B128 copies
// (no VGPR round-trip; ASYNCcnt-fenced).
// ---------------------------------------------------------------------------
__global__ __launch_bounds__(128)
void CAMD_kv_chunk_kernel(const float* __restrict__ K, const float* __restrict__ V,
                          float* __restrict__ KV)
{
  __shared__ __align__(16) float Ks[64 * 64];
  __shared__ __align__(16) float Vs[64 * 64];
  const int tid = threadIdx.x;
  const int c   = blockIdx.x;
  const float* Kc = K + (size_t)c * 64 * 64;
  const float* Vc = V + (size_t)c * 64 * 64;

  // 16B per lane per issue; 128 threads x 16B = 2KB per iteration, 8 iters each.
  for (int i = tid * 4; i < 64 * 64; i += 128 * 4) {
    async_ld_lds_b128((unsigned)(uintptr_t)(&Ks[i]), Kc + i);
    async_ld_lds_b128((unsigned)(uintptr_t)(&Vs[i]), Vc + i);
  }
  wait_asynccnt0();
  __syncthreads();

  const int wave = tid >> 5;  // k-dim tile
  const int lane = tid & 31;
  const int col  = lane & 15;
  const int hi   = lane >> 4;
  const int kr   = wave * 16 + col;  // A row = k-dim index

  #pragma unroll
  for (int vt = 0; vt < 4; ++vt) {
    v8f acc;
    #pragma unroll
    for (int v = 0; v < 8; ++v) acc[v] = 0.f;
    #pragma unroll
    for (int kc = 0; kc < 16; ++kc) {       // reduce over 64 time steps
      const int t0 = kc * 4 + hi * 2;
      v2f a, b;
      a[0] = Ks[(t0)     * 64 + kr];        // A[k-dim row][time] = K[t][k]
      a[1] = Ks[(t0 + 1) * 64 + kr];
      b[0] = Vs[(t0)     * 64 + vt * 16 + col];
      b[1] = Vs[(t0 + 1) * 64 + vt * 16 + col];
      acc = wmma_f32(a, b, acc);
    }
    #pragma unroll
    for (int v = 0; v < 8; ++v)
      KV[(size_t)c * 4096 + (size_t)(wave * 16 + v + 8 * hi) * 64 + vt * 16 + col] = acc[v];
  }
}

// ---------------------------------------------------------------------------
// Kernel C: in-place exclusive prefix scan over chunks (per (k,v) position).
// Writes total sum into extra slot nc.  4096 threads, <=128 serial steps.
// ---------------------------------------------------------------------------
__global__ __launch_bounds__(256)
void CAMD_scan_kernel(float* __restrict__ KV, int nc)
{
  const int p = blockIdx.x * 256 + threadIdx.x;  // 0..4095
  float run = 0.f;
  for (int c = 0; c < nc; ++c) {
    const float t = KV[(size_t)c * 4096 + p];
    KV[(size_t)c * 4096 + p] = run;
    run += t;
  }
  KV[(size_t)nc * 4096 + p] = run;
}

__global__ __launch_bounds__(256)
void CAMD_zero_kernel(float* __restrict__ p, int n)
{
  const int i = blockIdx.x * 256 + threadIdx.x;
  if (i < n) p[i] = 0.f;
}

// ---------------------------------------------------------------------------
// Kernel D: per 16-query tile, one modality:
//   out_tile += Q_tile @ P[C0]  (prefix state of fully-past chunks)
//            +  sum over boundary chunks of  mask(Q K_c^T) @ V_c
// blockDim = 32 (one wave per tile -> divergent chunk loops are safe)
// ---------------------------------------------------------------------------
__global__ __launch_bounds__(32)
void CAMD_attn_kernel(const float* __restrict__ Q, const float* __restrict__ Xref,
                      const float* __restrict__ Km, const float* __restrict__ Vm,
                      const float* __restrict__ Pm, const float* __restrict__ Xm,
                      int Tm, int dm, int nc, float* __restrict__ out)
{
  __shared__ float Sm[16 * 64];  // masked scores, C-layout -> A-layout staging

  const int lane = threadIdx.x;
  const int col  = lane & 15;
  const int hi   = lane >> 4;
  const int row0 = blockIdx.x * 16;

  // query times for the 8 accumulator rows this lane owns, plus tile bounds
  float t1r[8];
  #pragma unroll
  for (int v = 0; v < 8; ++v)
    t1r[v] = Xref[(size_t)(row0 + v + 8 * hi) * 64 + 63];
  const float tmin = Xref[(size_t)row0 * 64 + 63];
  const float tmax = Xref[(size_t)(row0 + 15) * 64 + 63];

  // upper_bound: count of key times <= x  (t2 is sorted)
  auto ub = [&](float x) {
    int lo = 0, hic = Tm;
    while (lo < hic) {
      const int mid = (lo + hic) >> 1;
      const float tv = Xm[(size_t)mid * dm + (dm - 1)];
      if (tv <= x) lo = mid + 1; else hic = mid;
    }
    return lo;
  };
  const int idx_lo = ub(tmin);
  const int idx_hi = ub(tmax);
  int C0 = idx_lo >> 6;
  if (C0 > nc) C0 = nc;

  v8f acc[4];
  #pragma unroll
  for (int vt = 0; vt < 4; ++vt)
    #pragma unroll
    for (int v = 0; v < 8; ++v) acc[vt][v] = 0.f;

  const float* Qrow = Q + (size_t)(row0 + col) * 64;

  // inter-chunk part: Q_tile (16x64) @ P[C0] (64x64)
  const float* Pc = Pm + (size_t)C0 * 4096;
  #pragma unroll
  for (int vt = 0; vt < 4; ++vt) {
    #pragma unroll
    for (int kc = 0; kc < 16; ++kc) {
      const int k0 = kc * 4 + hi * 2;
      v2f a, b;
      a[0] = Qrow[k0];
      a[1] = Qrow[k0 + 1];
      b[0] = Pc[(size_t)(k0)     * 64 + vt * 16 + col];
      b[1] = Pc[(size_t)(k0 + 1) * 64 + vt * 16 + col];
      acc[vt] = wmma_f32(a, b, acc[vt]);
    }
  }

  // boundary chunks: masked attention
  for (int c = C0; c < nc && c * 64 < idx_hi; ++c) {
    const float* Kc = Km + (size_t)c * 64 * 64;
    const float* Vc = Vm + (size_t)c * 64 * 64;

    // boundary chunks walk sequentially: prefetch next chunk (global_prefetch_b8)
    if (c + 1 < nc) {
      __builtin_prefetch(Km + (size_t)(c + 1) * 64 * 64, 0, 1);
      __builtin_prefetch(Vm + (size_t)(c + 1) * 64 * 64, 0, 1);
    }

    // scores S = Q_tile @ K_c^T   (16 queries x 64 keys), time-masked
    #pragma unroll
    for (int jt = 0; jt < 4; ++jt) {
      v8f s;
      #pragma unroll
      for (int v = 0; v < 8; ++v) s[v] = 0.f;
      #pragma unroll
      for (int kc = 0; kc < 16; ++kc) {
        const int k0 = kc * 4 + hi * 2;
        v2f a, b;
        a[0] = Qrow[k0];
        a[1] = Qrow[k0 + 1];
        b[0] = Kc[(size_t)(jt * 16 + col) * 64 + k0];      // B[k][key] = K[key][k]
        b[1] = Kc[(size_t)(jt * 16 + col) * 64 + k0 + 1];
        s = wmma_f32(a, b, s);
      }
      const int key = c * 64 + jt * 16 + col;
      const float t2v = Xm[(size_t)key * dm + (dm - 1)];
      #pragma unroll
      for (int v = 0; v < 8; ++v) {
        const float sv = (t2v <= t1r[v]) ? s[v] : 0.f;
        Sm[(v + 8 * hi) * 64 + jt * 16 + col] = sv;
      }
    }
    __syncthreads();

    // out += S (16x64) @ V_c (64x64)
    #pragma unroll
    for (int vt = 0; vt < 4; ++vt) {
      #pragma unroll
      for (int kc = 0; kc < 16; ++kc) {
        const int k0 = kc * 4 + hi * 2;
        v2f a, b;
        a[0] = Sm[col * 64 + k0];
        a[1] = Sm[col * 64 + k0 + 1];
        b[0] = Vc[(size_t)(k0)     * 64 + vt * 16 + col];
        b[1] = Vc[(size_t)(k0 + 1) * 64 + vt * 16 + col];
        acc[vt] = wmma_f32(a, b, acc[vt]);
      }
    }
    __syncthreads();
  }

  #pragma unroll
  for (int vt = 0; vt < 4; ++vt)
    #pragma unroll
    for (int v = 0; v < 8; ++v) {
      const size_t o = (size_t)(row0 + v + 8 * hi) * 64 + vt * 16 + col;
      out[o] += acc[vt][v];
    }
}

// ---------------------------------------------------------------------------
// Host launcher
// ---------------------------------------------------------------------------
extern "C" void kernel_launch(void* const* d_in, const int* in_sizes, int n_in,
                              void* d_out, int out_size, void* d_ws, size_t ws_size,
                              hipStream_t stream)
{
  (void)in_sizes; (void)n_in; (void)out_size; (void)ws_size;

  // MODS = {reference: d=64,T=8192}, {m1: d=32,T=8192}, {m2: d=48,T=6144}
  const int T_ref = 8192, T_m1 = 8192, T_m2 = 6144;
  const int d_ref = 64,   d_m1 = 32,   d_m2 = 48;
  const int nc_ref = T_ref / 64, nc_m1 = T_m1 / 64, nc_m2 = T_m2 / 64;

  const float* X_ref = (const float*)d_in[0];
  const float* X_m1  = (const float*)d_in[1];
  const float* X_m2  = (const float*)d_in[2];

  // params flattened in pytree (sorted-dict) order:
  // W_K{m1,m2,reference}, W_Q, W_V{m1,m2,reference}; each mlp = (w1,b1,w2,b2)
  const float* WK_m1[4]  = {(const float*)d_in[3],  (const float*)d_in[4],
                            (const float*)d_in[5],  (const float*)d_in[6]};
  const float* WK_m2[4]  = {(const float*)d_in[7],  (const float*)d_in[8],
                            (const float*)d_in[9],  (const float*)d_in[10]};
  const float* WK_ref[4] = {(const float*)d_in[11], (const float*)d_in[12],
                            (const float*)d_in[13], (const float*)d_in[14]};
  const float* WQ[4]     = {(const float*)d_in[15], (const float*)d_in[16],
                            (const float*)d_in[17], (const float*)d_in[18]};
  const float* WV_m1[4]  = {(const float*)d_in[19], (const float*)d_in[20],
                            (const float*)d_in[21], (const float*)d_in[22]};
  const float* WV_m2[4]  = {(const float*)d_in[23], (const float*)d_in[24],
                            (const float*)d_in[25], (const float*)d_in[26]};
  const float* WV_ref[4] = {(const float*)d_in[27], (const float*)d_in[28],
                            (const float*)d_in[29], (const float*)d_in[30]};

  // workspace carve-up (floats)
  float* ws = (float*)d_ws;
  size_t off = 0;
  float* Q      = ws + off; off += (size_t)T_ref * 64;
  float* K_ref  = ws + off; off += (size_t)T_ref * 64;
  float* V_ref  = ws + off; off += (size_t)T_ref * 64;
  float* KV_ref = ws + off; off += (size_t)(nc_ref + 1) * 4096;
  float* K_m1   = ws + off; off += (size_t)T_m1 * 64;
  float* V_m1   = ws + off; off += (size_t)T_m1 * 64;
  float* KV_m1  = ws + off; off += (size_t)(nc_m1 + 1) * 4096;
  float* K_m2   = ws + off; off += (size_t)T_m2 * 64;
  float* V_m2   = ws + off; off += (size_t)T_m2 * 64;
  float* KV_m2  = ws + off; off += (size_t)(nc_m2 + 1) * 4096;

  float* out = (float*)d_out;

  // 1) MLPs: Q (feature-mapped), K_m (feature-mapped), V_m (plain)
  CAMD_mlp_kernel<<<T_ref / 64, 128, 0, stream>>>(X_ref, d_ref, WQ[0], WQ[1], WQ[2], WQ[3], Q, 1);
  CAMD_mlp_kernel<<<T_ref / 64, 128, 0, stream>>>(X_ref, d_ref, WK_ref[0], WK_ref[1], WK_ref[2], WK_ref[3], K_ref, 1);
  CAMD_mlp_kernel<<<T_ref / 64, 128, 0, stream>>>(X_ref, d_ref, WV_ref[0], WV_ref[1], WV_ref[2], WV_ref[3], V_ref, 0);
  CAMD_mlp_kernel<<<T_m1 / 64, 128, 0, stream>>>(X_m1, d_m1, WK_m1[0], WK_m1[1], WK_m1[2], WK_m1[3], K_m1, 1);
  CAMD_mlp_kernel<<<T_m1 / 64, 128, 0, stream>>>(X_m1, d_m1, WV_m1[0], WV_m1[1], WV_m1[2], WV_m1[3], V_m1, 0);
  CAMD_mlp_kernel<<<T_m2 / 64, 128, 0, stream>>>(X_m2, d_m2, WK_m2[0], WK_m2[1], WK_m2[2], WK_m2[3], K_m2, 1);
  CAMD_mlp_kernel<<<T_m2 / 64, 128, 0, stream>>>(X_m2, d_m2, WV_m2[0], WV_m2[1], WV_m2[2], WV_m2[3], V_m2, 0);

  // 2) per-chunk KV states (async global->LDS staging inside)
  CAMD_kv_chunk_kernel<<<nc_ref, 128, 0, stream>>>(K_ref, V_ref, KV_ref);
  CAMD_kv_chunk_kernel<<<nc_m1, 128, 0, stream>>>(K_m1, V_m1, KV_m1);
  CAMD_kv_chunk_kernel<<<nc_m2, 128, 0, stream>>>(K_m2, V_m2, KV_m2);

  // 3) exclusive prefix scan over chunks (in place, plus total slot)
  CAMD_scan_kernel<<<16, 256, 0, stream>>>(KV_ref, nc_ref);
  CAMD_scan_kernel<<<16, 256, 0, stream>>>(KV_m1, nc_m1);
  CAMD_scan_kernel<<<16, 256, 0, stream>>>(KV_m2, nc_m2);

  // 4) output: zero, then accumulate the three modalities
  const int on = T_ref * 64;
  CAMD_zero_kernel<<<(on + 255) / 256, 256, 0, stream>>>(out, on);
  CAMD_attn_kernel<<<T_ref / 16, 32, 0, stream>>>(Q, X_ref, K_ref, V_ref, KV_ref, X_ref, T_ref, d_ref, nc_ref, out);
  CAMD_attn_kernel<<<T_ref / 16, 32, 0, stream>>>(Q, X_ref, K_m1, V_m1, KV_m1, X_m1, T_m1, d_m1, nc_m1, out);
  CAMD_attn_kernel<<<T_ref / 16, 32, 0, stream>>>(Q, X_ref, K_m2, V_m2, KV_m2, X_m2, T_m2, d_m2, nc_m2, out);
}